// ELBO_14585708937856
// MI455X (gfx1250) — compile-verified
//
#include <hip/hip_runtime.h>
#include <math.h>

// ---------------------------------------------------------------------------
// ELBO reduction for diagonal-covariance MVN log-prob (T=128, B=4096, NOBS=32)
//   out = kl + 0.5*(N/B) * ( sum_{t,b,k} diff^2/var[k]  +  T*B*(logdet + 32*log2pi) )
// Memory-bound: 134 MB streamed @ 23.3 TB/s  ->  ~5.8 us floor (L2-resident on
// replay: 134 MB < 192 MB L2, so default RT temporal hints are correct).
// WMMA (V_WMMA_F32_16X16X4_F32, exact f32) performs the cross-lane reduce.
// ---------------------------------------------------------------------------

typedef __attribute__((ext_vector_type(2))) float v2f;
typedef __attribute__((ext_vector_type(8))) float v8f;

#define T_DIM   128
#define B_DIM   4096
#define NOBS    32
#define LOG_2PI 1.8378770664093453f

#define BLOCKS  1024
#define THREADS 256
#define WAVES_PER_BLOCK (THREADS / 32)

// Exact compile-time trip structure: Q float4s split evenly over the grid.
#define QF4     ((T_DIM * B_DIM * NOBS) / 4)   // 4,194,304 float4s
#define STRIDE  (BLOCKS * THREADS)             // 262,144 threads
#define ITERS   (QF4 / STRIDE)                 // 16, no remainder

__device__ __forceinline__ float softplus_f(float x) {
    // numerically stable softplus: max(x,0) + log1p(exp(-|x|))
    return fmaxf(x, 0.0f) + log1pf(expf(-fabsf(x)));
}

// Reduce one f32 value per lane (32 lanes) to the full wave sum, exactly in
// f32, using V_WMMA_F32_16X16X4_F32:
//   A (16x4): each lane's acc appears in exactly one slot (A vgpr0 = acc,
//             A vgpr1 = 0), B = all-ones, C = 0.
//   => each column of D holds the 16 row-sums; summing a lane's 8 D regs
//      gives rows 0-7 (lanes 0-15) / rows 8-15 (lanes 16-31), so
//      lane0 + lane16 cover all 32 accumulators. Exact f32, deterministic.
// EXEC must be all ones at the call site (callers are fully convergent).
__device__ __forceinline__ float wave_reduce_wmma(float acc) {
    v2f a; a[0] = acc;  a[1] = 0.0f;
    v2f b; b[0] = 1.0f; b[1] = 1.0f;
    v8f c = {};
    v8f d = __builtin_amdgcn_wmma_f32_16x16x4_f32(
        /*neg_a=*/false, a, /*neg_b=*/false, b,
        /*c_mod=*/(short)0, c, /*reuse_a=*/false, /*reuse_b=*/false);
    float t = d[0] + d[1] + d[2] + d[3] + d[4] + d[5] + d[6] + d[7];
    return __shfl(t, 0, 32) + __shfl(t, 16, 32);   // broadcast to all lanes
}

// Pass 1: streaming Mahalanobis partial sums, one f32 partial per block.
__global__ void __launch_bounds__(THREADS)
elbo_partial(const float* __restrict__ Y,
             const float* __restrict__ Yhat,
             const float* __restrict__ raw_sn,
             float* __restrict__ partials) {
    const int tid = threadIdx.x;
    const int gid = blockIdx.x * THREADS + tid;

    // STRIDE*4 is a multiple of NOBS=32 -> this thread's obs-dim phase is
    // loop-invariant; compute its 4 inverse variances once.
    const int k0 = (4 * gid) & (NOBS - 1);
    const float ivx = 1.0f / softplus_f(raw_sn[k0 + 0]);
    const float ivy = 1.0f / softplus_f(raw_sn[k0 + 1]);
    const float ivz = 1.0f / softplus_f(raw_sn[k0 + 2]);
    const float ivw = 1.0f / softplus_f(raw_sn[k0 + 3]);

    const float4* __restrict__ Y4  = (const float4*)Y;
    const float4* __restrict__ Yh4 = (const float4*)Yhat;

    // 4 independent accumulators; unroll-by-4 keeps 8 B128 loads in flight
    // per unrolled body without serializing the FMA chains.
    float acc0 = 0.0f, acc1 = 0.0f, acc2 = 0.0f, acc3 = 0.0f;
#pragma unroll 4
    for (int it = 0; it < ITERS; ++it) {
        const int i = gid + it * STRIDE;                 // global_load_b128 x2
        float4 y = Y4[i];
        float4 h = Yh4[i];
        float dx = h.x - y.x;
        float dy = h.y - y.y;
        float dz = h.z - y.z;
        float dw = h.w - y.w;
        acc0 = fmaf(dx * dx, ivx, acc0);
        acc1 = fmaf(dy * dy, ivy, acc1);
        acc2 = fmaf(dz * dz, ivz, acc2);
        acc3 = fmaf(dw * dw, ivw, acc3);
    }
    float acc = (acc0 + acc1) + (acc2 + acc3);

    // Fully convergent here: WMMA wave reduction (EXEC all ones).
    float wsum = wave_reduce_wmma(acc);

    __shared__ float lds[WAVES_PER_BLOCK];
    const int wave = tid >> 5;
    const int lane = tid & 31;
    if (lane == 0) lds[wave] = wsum;
    __syncthreads();
    if (tid == 0) {
        float s = 0.0f;
#pragma unroll
        for (int w = 0; w < WAVES_PER_BLOCK; ++w) s += lds[w];   // fixed order
        partials[blockIdx.x] = s;
    }
}

// Pass 2: one wave reduces the 1024 block partials (fixed order -> exact
// determinism), then finalizes logdet / constants / kl / N scaling.
__global__ void __launch_bounds__(32)
elbo_final(const float* __restrict__ partials,
           const float* __restrict__ raw_sn,
           const float* __restrict__ kl,
           const int*   __restrict__ Np,
           float* __restrict__ out) {
    const int lane = threadIdx.x;
    float acc = 0.0f;
    for (int j = lane; j < BLOCKS; j += 32) acc += partials[j];

    float S = wave_reduce_wmma(acc);                     // sum of all maha

    if (lane == 0) {
        float logdet = 0.0f;
        for (int k = 0; k < NOBS; ++k)
            logdet += logf(softplus_f(raw_sn[k]));
        const float C     = logdet + (float)NOBS * LOG_2PI;
        const float scale = 0.5f * ((float)Np[0] / (float)B_DIM);
        out[0] = scale * (S + (float)(T_DIM * B_DIM) * C) + kl[0];
    }
}

extern "C" void kernel_launch(void* const* d_in, const int* in_sizes, int n_in,
                              void* d_out, int out_size, void* d_ws, size_t ws_size,
                              hipStream_t stream) {
    (void)in_sizes; (void)n_in; (void)out_size; (void)ws_size;

    const float* Y      = (const float*)d_in[0];
    const float* Yhat   = (const float*)d_in[1];
    const float* raw_sn = (const float*)d_in[2];
    const float* kl     = (const float*)d_in[3];
    const int*   N      = (const int*)d_in[4];

    float* partials = (float*)d_ws;                      // BLOCKS floats

    elbo_partial<<<BLOCKS, THREADS, 0, stream>>>(Y, Yhat, raw_sn, partials);
    elbo_final<<<1, 32, 0, stream>>>(partials, raw_sn, kl, N, (float*)d_out);
}